// GIN_36335423324412
// MI455X (gfx1250) — compile-verified
//
#include <hip/hip_runtime.h>
#include <hip/hip_bf16.h>

typedef __attribute__((ext_vector_type(2))) float v2f;
typedef __attribute__((ext_vector_type(8))) float v8f;

#define F_IN 128
#define HID  16
#define N_CLS 40

// ---------------------------------------------------------------------------
// Phase A1: per-destination degree count. 1.6M u32 atomics, L2-resident.
// ---------------------------------------------------------------------------
__global__ void __launch_bounds__(256)
gin_deg_count(const long long* __restrict__ dst, int* __restrict__ deg, int n_edges) {
  const int e = blockIdx.x * blockDim.x + threadIdx.x;
  if (e >= n_edges) return;
  atomicAdd(&deg[(int)dst[e]], 1);
}

// ---------------------------------------------------------------------------
// Phase A2: contiguous region allocation per node, wave-aggregated:
// wave32 prefix-sum of degrees via __shfl_up, one atomicAdd per wave on a
// global cursor (~3.1K contended atomics total instead of a 100K-wide scan).
// Regions are contiguous per node (id order irrelevant for correctness).
// Writes offs[] and a working copy cur[] for the fill pass.
// ---------------------------------------------------------------------------
__global__ void __launch_bounds__(256)
gin_alloc_offsets(const int* __restrict__ deg, int* __restrict__ offs,
                  int* __restrict__ cur, int* __restrict__ cursor, int n_nodes) {
  const int node = blockIdx.x * blockDim.x + threadIdx.x;
  const int lane = threadIdx.x & 31;
  const int d = (node < n_nodes) ? deg[node] : 0;

  // inclusive scan within wave32
  int s = d;
#pragma unroll
  for (int off = 1; off < 32; off <<= 1) {
    const int t = __shfl_up(s, off, 32);
    if (lane >= off) s += t;
  }
  const int total = __shfl(s, 31, 32);
  int base = 0;
  if (lane == 0 && total > 0) base = atomicAdd(cursor, total);
  base = __shfl(base, 0, 32);

  if (node < n_nodes) {
    const int o = base + s - d;   // exclusive position
    offs[node] = o;
    cur[node]  = o;
  }
}

// ---------------------------------------------------------------------------
// Phase A3: bucket fill: ebuf[cur[dst]++] = src. 1.6M u32 atomics.
// ---------------------------------------------------------------------------
__global__ void __launch_bounds__(256)
gin_fill(const long long* __restrict__ src, const long long* __restrict__ dst,
         int* __restrict__ cur, int* __restrict__ ebuf, int n_edges) {
  const int e = blockIdx.x * blockDim.x + threadIdx.x;
  if (e >= n_edges) return;
  const int d = (int)dst[e];
  const int p = atomicAdd(&cur[d], 1);
  ebuf[p] = (int)src[e];
}

// ---------------------------------------------------------------------------
// Kernel 1: y1 = x @ w1 (N x 128 @ 128 x 16) via V_WMMA_F32_16X16X4_F32.
// One wave32 per 16-row tile; 32 WMMA steps over K=128.
// A layout (16x4 f32): lanes 0-15 -> K={k,k+1}; lanes 16-31 -> K={k+2,k+3}.
// C/D layout: VGPR r holds row r (lanes 0-15) / row r+8 (lanes 16-31).
// ---------------------------------------------------------------------------
__global__ void __launch_bounds__(32)
gin_gemm1_wmma(const float* __restrict__ x, const float* __restrict__ w1,
               float* __restrict__ y1, int n_nodes) {
  const int lane = threadIdx.x;
  const int half = lane >> 4;
  const int l15  = lane & 15;
  const int tile = blockIdx.x;

  int rowA = tile * 16 + l15;
  if (rowA >= n_nodes) rowA = n_nodes - 1;   // clamp loads; EXEC stays all-1s
  const float* __restrict__ xr = x + (size_t)rowA * F_IN;
  const int kOff = half * 2;

  v8f acc = {};
#pragma unroll
  for (int k = 0; k < F_IN; k += 4) {
    v2f a, b;
    a.x = xr[k + kOff];
    a.y = xr[k + kOff + 1];
    b.x = w1[(k + kOff) * HID + l15];
    b.y = w1[(k + kOff + 1) * HID + l15];
    acc = __builtin_amdgcn_wmma_f32_16x16x4_f32(
        false, a, false, b, (short)0, acc, false, false);
  }

#pragma unroll
  for (int r = 0; r < 8; ++r) {
    const int row = tile * 16 + r + half * 8;
    if (row < n_nodes) y1[(size_t)row * HID + l15] = acc[r];
  }
}

// ---------------------------------------------------------------------------
// Aggregation (gather form): out[i] = op( feat[i] + bias + sum_{j in list(i)} feat[j] ).
// 4 lanes per node, float4 per lane (64B per neighbor row, L2-resident).
// No f32 atomics, deterministic per-node summation order given the list.
// relu_flag: 1 for layer 1 (bias = b1), 0 for layer 2 (bias = nullptr).
// ---------------------------------------------------------------------------
__global__ void __launch_bounds__(256)
gin_aggregate(const float* __restrict__ feat, const int* __restrict__ offs,
              const int* __restrict__ deg, const int* __restrict__ ebuf,
              const float* __restrict__ bias, float* __restrict__ out,
              int n_nodes, int relu_flag) {
  const int t = blockIdx.x * blockDim.x + threadIdx.x;
  const int node = t >> 2;
  const int q = (t & 3) * 4;           // feature quad
  if (node >= n_nodes) return;

  float4 acc = *(const float4*)(feat + (size_t)node * HID + q);
  if (bias) {
    const float4 bv = *(const float4*)(bias + q);
    acc.x += bv.x; acc.y += bv.y; acc.z += bv.z; acc.w += bv.w;
  }

  const int beg = offs[node];
  const int end = beg + deg[node];
  for (int e = beg; e < end; ++e) {
    const int s = ebuf[e];             // broadcast across the 4 lanes
    const float4 v = *(const float4*)(feat + (size_t)s * HID + q);
    acc.x += v.x; acc.y += v.y; acc.z += v.z; acc.w += v.w;
  }

  if (relu_flag) {
    acc.x = fmaxf(acc.x, 0.f); acc.y = fmaxf(acc.y, 0.f);
    acc.z = fmaxf(acc.z, 0.f); acc.w = fmaxf(acc.w, 0.f);
  }
  *(float4*)(out + (size_t)node * HID + q) = acc;
}

// ---------------------------------------------------------------------------
// Kernel 5: out = log_softmax(relu(s @ w2 + b2)) per row. w2/b2 in LDS.
// 640 FMAs per row is negligible; output 16 MB streamed to HBM.
// ---------------------------------------------------------------------------
__global__ void __launch_bounds__(256)
gin_gemm2_lsm(const float* __restrict__ s, const float* __restrict__ w2,
              const float* __restrict__ b2, float* __restrict__ out, int n_nodes) {
  __shared__ float sw[HID * N_CLS];
  __shared__ float sb[N_CLS];
  for (int i = threadIdx.x; i < HID * N_CLS; i += blockDim.x) sw[i] = w2[i];
  for (int i = threadIdx.x; i < N_CLS; i += blockDim.x) sb[i] = b2[i];
  __syncthreads();

  const int row = blockIdx.x * blockDim.x + threadIdx.x;
  if (row >= n_nodes) return;

  float sv[HID];
  const float4* sp = (const float4*)(s + (size_t)row * HID);
#pragma unroll
  for (int q = 0; q < 4; ++q) {
    const float4 a = sp[q];
    sv[q * 4 + 0] = a.x; sv[q * 4 + 1] = a.y;
    sv[q * 4 + 2] = a.z; sv[q * 4 + 3] = a.w;
  }

  float vals[N_CLS];
  float m = -INFINITY;
#pragma unroll
  for (int c = 0; c < N_CLS; ++c) {
    float a = sb[c];
#pragma unroll
    for (int k = 0; k < HID; ++k) a = fmaf(sv[k], sw[k * N_CLS + c], a);
    a = fmaxf(a, 0.f);
    vals[c] = a;
    m = fmaxf(m, a);
  }
  float sum = 0.f;
#pragma unroll
  for (int c = 0; c < N_CLS; ++c) sum += expf(vals[c] - m);
  const float lse = logf(sum) + m;

  float* op = out + (size_t)row * N_CLS;
#pragma unroll
  for (int c = 0; c < N_CLS; ++c) op[c] = vals[c] - lse;
}

// ---------------------------------------------------------------------------
// Launcher. Inputs: x, w1, b1, w2, b2, edge_index(int64, [2, E]).
// Workspace layout (ints first, then float buffers):
//   deg[N] | offs[N] | cur[N] | cursor[1 (+pad 15)] | ebuf[E] | bufA[N*16] | bufB[N*16]
// bufA = y1 then s2 (reuse); bufB = h1.
// ---------------------------------------------------------------------------
extern "C" void kernel_launch(void* const* d_in, const int* in_sizes, int n_in,
                              void* d_out, int out_size, void* d_ws, size_t ws_size,
                              hipStream_t stream) {
  const float*     x  = (const float*)d_in[0];
  const float*     w1 = (const float*)d_in[1];
  const float*     b1 = (const float*)d_in[2];
  const float*     w2 = (const float*)d_in[3];
  const float*     b2 = (const float*)d_in[4];
  const long long* ei = (const long long*)d_in[5];

  const int n_nodes = in_sizes[0] / F_IN;   // 100000
  const int n_edges = in_sizes[5] / 2;      // 1600000
  const long long* e_src = ei;
  const long long* e_dst = ei + n_edges;

  int* deg    = (int*)d_ws;
  int* offs   = deg  + n_nodes;
  int* cur    = offs + n_nodes;
  int* cursor = cur  + n_nodes;             // 16 ints reserved
  int* ebuf   = cursor + 16;
  float* bufA = (float*)(ebuf + n_edges);   // y1, later s2
  float* bufB = bufA + (size_t)n_nodes * HID;  // h1
  float* outp = (float*)d_out;

  const int eg = (n_edges + 255) / 256;
  const int ng = (n_nodes + 255) / 256;
  const int ag = (n_nodes * 4 + 255) / 256;

  // --- Build per-destination adjacency (reused by both layers) ---
  hipMemsetAsync(deg, 0, (size_t)n_nodes * sizeof(int), stream);
  hipMemsetAsync(cursor, 0, 16 * sizeof(int), stream);
  gin_deg_count<<<eg, 256, 0, stream>>>(e_dst, deg, n_edges);
  gin_alloc_offsets<<<ng, 256, 0, stream>>>(deg, offs, cur, cursor, n_nodes);
  gin_fill<<<eg, 256, 0, stream>>>(e_src, e_dst, cur, ebuf, n_edges);

  // --- Layer 1: y1 = x@w1 (WMMA), h1 = relu(y1 + gather-sum + b1) ---
  const int tiles = (n_nodes + 15) / 16;
  gin_gemm1_wmma<<<tiles, 32, 0, stream>>>(x, w1, bufA, n_nodes);
  gin_aggregate<<<ag, 256, 0, stream>>>(bufA, offs, deg, ebuf, b1, bufB,
                                        n_nodes, /*relu=*/1);

  // --- Layer 2: s2 = h1 + gather-sum(h1) ; out = log_softmax(relu(s2@w2+b2)) ---
  gin_aggregate<<<ag, 256, 0, stream>>>(bufB, offs, deg, ebuf, nullptr, bufA,
                                        n_nodes, /*relu=*/0);
  gin_gemm2_lsm<<<ng, 256, 0, stream>>>(bufA, w2, b2, outp, n_nodes);
}